// TGN_68985764708362
// MI455X (gfx1250) — compile-verified
//
#include <hip/hip_runtime.h>
#include <math.h>

// TGN memory update for gfx1250 (MI455X), fp32 WMMA path, B-fragment reuse x2.
//
// Phase 1: scatter-max of packed (time,index) keys -> latest event per node.
// Phase 2: per wave, TWO 16-node tiles share every weight (B) fragment:
//          each k-step loads 3 B-frags (r/z/n rows) and issues 6
//          V_WMMA_F32_16X16X4_F32, halving L2 weight traffic per FLOP.
//          A-frags for both tiles come back in one ds_load_2addr_stride64_b64.
// LDS: 2 x (16 x 512) f32 tiles, XOR bank swizzle col^=4*row -> exactly 64KB,
//      conflict-free A-fragment reads (all 64 banks hit once per access).
//      Swizzle constant is a multiple of 4 -> 4-aligned column quads stay
//      contiguous, so the tile build uses b128 global loads + b128 LDS stores.
//      Build loop kept rolled (#pragma unroll 1) to limit I$ footprint.

#define D_DIM 128
#define M_DIM 512              // 2D + T + F
#define TILE_F (16 * M_DIM)    // floats per X tile

typedef __attribute__((ext_vector_type(2))) float v2f;
typedef __attribute__((ext_vector_type(4))) float v4f;
typedef __attribute__((ext_vector_type(8))) float v8f;

__global__ void tgn_init_keys(unsigned long long* __restrict__ keys, int N) {
    int i = blockIdx.x * blockDim.x + threadIdx.x;
    if (i < N) keys[i] = 0ull;
}

// key = ((float_bits(t)+1) << 32) | combined_index ; t >= 0 so float bits are
// monotone as uint; +1 guarantees key != 0 for any real event. atomic max
// reproduces the reference's stable-sort + segment_max tie-break exactly.
__global__ void tgn_scatter_max(const int* __restrict__ src,
                                const int* __restrict__ dst,
                                const float* __restrict__ t,
                                unsigned long long* __restrict__ keys, int E) {
    int e = blockIdx.x * blockDim.x + threadIdx.x;
    if (e >= 2 * E) return;
    int ev = (e < E) ? e : e - E;
    int node = (e < E) ? src[e] : dst[ev];
    unsigned int tb = __float_as_uint(t[ev]) + 1u;
    unsigned long long key = ((unsigned long long)tb << 32) | (unsigned int)e;
    atomicMax(&keys[node], key);
}

__launch_bounds__(32)
__global__ void tgn_gru_wmma(const float* __restrict__ memory,
                             const float* __restrict__ last_update,
                             const float* __restrict__ t,
                             const float* __restrict__ feat,
                             const float* __restrict__ time_w,
                             const float* __restrict__ time_b,
                             const float* __restrict__ w_ih,
                             const float* __restrict__ w_hh,
                             const float* __restrict__ b_ih,
                             const float* __restrict__ b_hh,
                             const int* __restrict__ src,
                             const int* __restrict__ dst,
                             const unsigned long long* __restrict__ keys,
                             float* __restrict__ out_mem,
                             float* __restrict__ out_lu,
                             int N, int E) {
    __shared__ float X[2 * TILE_F];     // 65536 bytes exactly

    const int lane = threadIdx.x;       // wave32: one wave per workgroup
    const int base = blockIdx.x * 32;   // 32 nodes (2 tiles) per block
    const int k4 = lane * 4;            // this lane's column quad (0..124)

    const v4f zero4 = {0.f, 0.f, 0.f, 0.f};
    const v4f tw4 = *(const v4f*)&time_w[k4];
    const v4f tb4 = *(const v4f*)&time_b[k4];

    // ---- build both message tiles; X[:, 0:128] == H (memory[node]) ----
    // swizzled store index within a tile: r*512 + (k ^ (4*r)); quads stay
    // contiguous because the swizzle constant is a multiple of 4.
    unsigned hasmask = 0;               // uniform across lanes
#pragma unroll 1
    for (int m = 0; m < 32; ++m) {
        int node = base + m;            // uniform across lanes
        bool valid = node < N;
        unsigned long long key = valid ? keys[node] : 0ull;
        int has = (key != 0ull) ? 1 : 0;
        if (valid) hasmask |= ((unsigned)has << m);
        unsigned int e = (unsigned int)(key & 0xffffffffu);
        int ev = 0, other = 0;
        if (has) {
            if (e < (unsigned)E) { ev = (int)e;       other = dst[ev]; }
            else                 { ev = (int)(e - E); other = src[ev]; }
        }
        float* row = &X[(m >> 4) * TILE_F + (m & 15) * M_DIM];
        const int sw = (m & 15) << 2;   // XOR swizzle constant for this row
        if (!valid) {
            *(v4f*)&row[k4 ^ sw]                 = zero4;
            *(v4f*)&row[(D_DIM + k4) ^ sw]       = zero4;
            *(v4f*)&row[(2 * D_DIM + k4) ^ sw]   = zero4;
            *(v4f*)&row[(3 * D_DIM + k4) ^ sw]   = zero4;
            continue;
        }
        float dt = has ? (t[ev] - last_update[node]) : 0.0f;
        v4f hq = *(const v4f*)&memory[(size_t)node * D_DIM + k4];
        v4f oq = has ? *(const v4f*)&memory[(size_t)other * D_DIM + k4] : zero4;
        v4f fq = has ? *(const v4f*)&feat[(size_t)ev * D_DIM + k4] : zero4;
        v4f tq;
        if (has) {
            tq.x = __cosf(dt * tw4.x + tb4.x);
            tq.y = __cosf(dt * tw4.y + tb4.y);
            tq.z = __cosf(dt * tw4.z + tb4.z);
            tq.w = __cosf(dt * tw4.w + tb4.w);
        } else {
            tq = zero4;
        }
        *(v4f*)&row[k4 ^ sw]               = hq;  // self memory == H row
        *(v4f*)&row[(D_DIM + k4) ^ sw]     = oq;
        *(v4f*)&row[(2 * D_DIM + k4) ^ sw] = tq;
        *(v4f*)&row[(3 * D_DIM + k4) ^ sw] = fq;
    }
    __syncthreads();   // single-wave WG: lowers to s_nop; compiler LDS fence

    const int lo = lane & 15;           // A row / B column for this lane
    const int hi = lane >> 4;           // K half-select
    const int xc = lo << 2;             // per-lane A-row swizzle constant
    const int klane = 2 * hi;
    const float* A0 = &X[0 * TILE_F + lo * M_DIM];
    const float* A1 = &X[1 * TILE_F + lo * M_DIM];
    const v8f z8 = {0.f, 0.f, 0.f, 0.f, 0.f, 0.f, 0.f, 0.f};

#pragma unroll 1
    for (int c = 0; c < 8; ++c) {       // 8 column-tiles of D = 128
        v8f ai_r0 = z8, ai_z0 = z8, ai_n0 = z8;   // tile 0: x @ W_ih^T
        v8f ah_r0 = z8, ah_z0 = z8, ah_n0 = z8;   // tile 0: h @ W_hh^T
        v8f ai_r1 = z8, ai_z1 = z8, ai_n1 = z8;   // tile 1
        v8f ah_r1 = z8, ah_z1 = z8, ah_n1 = z8;

        const float* Bi_r = w_ih + (size_t)(0 * 128 + c * 16 + lo) * M_DIM;
        const float* Bi_z = w_ih + (size_t)(1 * 128 + c * 16 + lo) * M_DIM;
        const float* Bi_n = w_ih + (size_t)(2 * 128 + c * 16 + lo) * M_DIM;
        const float* Bh_r = w_hh + (size_t)(0 * 128 + c * 16 + lo) * D_DIM;
        const float* Bh_z = w_hh + (size_t)(1 * 128 + c * 16 + lo) * D_DIM;
        const float* Bh_n = w_hh + (size_t)(2 * 128 + c * 16 + lo) * D_DIM;

        // input-hidden GEMM: K = 512; each B-frag feeds both node tiles
        for (int k0 = 0; k0 < M_DIM; k0 += 4) {
            int k = k0 + klane;
            int kc = k ^ xc;
            v2f a0 = *(const v2f*)&A0[kc];
            v2f a1 = *(const v2f*)&A1[kc];
            v2f br = *(const v2f*)&Bi_r[k];
            v2f bz = *(const v2f*)&Bi_z[k];
            v2f bn = *(const v2f*)&Bi_n[k];
            ai_r0 = __builtin_amdgcn_wmma_f32_16x16x4_f32(
                false, a0, false, br, (short)0, ai_r0, false, false);
            ai_r1 = __builtin_amdgcn_wmma_f32_16x16x4_f32(
                false, a1, false, br, (short)0, ai_r1, false, false);
            ai_z0 = __builtin_amdgcn_wmma_f32_16x16x4_f32(
                false, a0, false, bz, (short)0, ai_z0, false, false);
            ai_z1 = __builtin_amdgcn_wmma_f32_16x16x4_f32(
                false, a1, false, bz, (short)0, ai_z1, false, false);
            ai_n0 = __builtin_amdgcn_wmma_f32_16x16x4_f32(
                false, a0, false, bn, (short)0, ai_n0, false, false);
            ai_n1 = __builtin_amdgcn_wmma_f32_16x16x4_f32(
                false, a1, false, bn, (short)0, ai_n1, false, false);
        }
        // hidden-hidden GEMM: K = 128, A read from X[:, 0:128] (== H)
        for (int k0 = 0; k0 < D_DIM; k0 += 4) {
            int k = k0 + klane;
            int kc = k ^ xc;
            v2f a0 = *(const v2f*)&A0[kc];
            v2f a1 = *(const v2f*)&A1[kc];
            v2f br = *(const v2f*)&Bh_r[k];
            v2f bz = *(const v2f*)&Bh_z[k];
            v2f bn = *(const v2f*)&Bh_n[k];
            ah_r0 = __builtin_amdgcn_wmma_f32_16x16x4_f32(
                false, a0, false, br, (short)0, ah_r0, false, false);
            ah_r1 = __builtin_amdgcn_wmma_f32_16x16x4_f32(
                false, a1, false, br, (short)0, ah_r1, false, false);
            ah_z0 = __builtin_amdgcn_wmma_f32_16x16x4_f32(
                false, a0, false, bz, (short)0, ah_z0, false, false);
            ah_z1 = __builtin_amdgcn_wmma_f32_16x16x4_f32(
                false, a1, false, bz, (short)0, ah_z1, false, false);
            ah_n0 = __builtin_amdgcn_wmma_f32_16x16x4_f32(
                false, a0, false, bn, (short)0, ah_n0, false, false);
            ah_n1 = __builtin_amdgcn_wmma_f32_16x16x4_f32(
                false, a1, false, bn, (short)0, ah_n1, false, false);
        }

        const int col = c * 16 + lo;
        const float brz = b_ih[col] + b_hh[col];              // r gate bias
        const float bzz = b_ih[128 + col] + b_hh[128 + col];  // z gate bias
        const float bin = b_ih[256 + col];
        const float bhn = b_hh[256 + col];

        for (int i = 0; i < 8; ++i) {
            const int m = i + 8 * hi;            // row within tile
            // ---- tile 0 ----
            {
                int node = base + m;
                float hval = X[0 * TILE_F + m * M_DIM + (col ^ (m << 2))];
                float r = 1.0f / (1.0f + __expf(-(ai_r0[i] + ah_r0[i] + brz)));
                float z = 1.0f / (1.0f + __expf(-(ai_z0[i] + ah_z0[i] + bzz)));
                float n = tanhf(ai_n0[i] + bin + r * (ah_n0[i] + bhn));
                float newh = (1.0f - z) * n + z * hval;
                if (node < N)
                    out_mem[(size_t)node * D_DIM + col] =
                        ((hasmask >> m) & 1u) ? newh : hval;
            }
            // ---- tile 1 ----
            {
                int node = base + 16 + m;
                float hval = X[1 * TILE_F + m * M_DIM + (col ^ (m << 2))];
                float r = 1.0f / (1.0f + __expf(-(ai_r1[i] + ah_r1[i] + brz)));
                float z = 1.0f / (1.0f + __expf(-(ai_z1[i] + ah_z1[i] + bzz)));
                float n = tanhf(ai_n1[i] + bin + r * (ah_n1[i] + bhn));
                float newh = (1.0f - z) * n + z * hval;
                if (node < N)
                    out_mem[(size_t)node * D_DIM + col] =
                        ((hasmask >> (16 + m)) & 1u) ? newh : hval;
            }
        }
    }

    // ---- new_last_update: each lane owns one of the 32 nodes ----
    {
        int node = base + lane;
        if (node < N) {
            unsigned long long key = keys[node];
            if (key != 0ull) {
                unsigned int e = (unsigned int)(key & 0xffffffffu);
                int ev = (e < (unsigned)E) ? (int)e : (int)(e - E);
                out_lu[node] = t[ev];
            } else {
                out_lu[node] = last_update[node];
            }
        }
    }
}

extern "C" void kernel_launch(void* const* d_in, const int* in_sizes, int n_in,
                              void* d_out, int out_size, void* d_ws, size_t ws_size,
                              hipStream_t stream) {
    const float* memory      = (const float*)d_in[0];
    const float* last_update = (const float*)d_in[1];
    const float* t           = (const float*)d_in[2];
    const float* feat        = (const float*)d_in[3];
    const float* time_w      = (const float*)d_in[4];
    const float* time_b      = (const float*)d_in[5];
    const float* w_ih        = (const float*)d_in[6];
    const float* w_hh        = (const float*)d_in[7];
    const float* b_ih        = (const float*)d_in[8];
    const float* b_hh        = (const float*)d_in[9];
    const int*   src         = (const int*)d_in[10];
    const int*   dst         = (const int*)d_in[11];

    const int N = in_sizes[1];   // last_update: [N]
    const int E = in_sizes[2];   // t: [E]

    float* out_mem = (float*)d_out;
    float* out_lu  = (float*)d_out + (size_t)N * D_DIM;

    unsigned long long* keys = (unsigned long long*)d_ws;  // N * 8 bytes

    tgn_init_keys<<<(N + 255) / 256, 256, 0, stream>>>(keys, N);
    tgn_scatter_max<<<(2 * E + 255) / 256, 256, 0, stream>>>(src, dst, t, keys, E);
    tgn_gru_wmma<<<(N + 31) / 32, 32, 0, stream>>>(
        memory, last_update, t, feat, time_w, time_b,
        w_ih, w_hh, b_ih, b_hh, src, dst, keys, out_mem, out_lu, N, E);
}